// ROLAND_52132313039364
// MI455X (gfx1250) — compile-verified
//
#include <hip/hip_runtime.h>
#include <math.h>

#define N_NODES 50000
#define N_EDGES 800000
#define DIM     128
#define N_LAYERS 2
#define N_EL    100000
#define BN_PB   250   // partial-reduction blocks for BN stats

typedef float v2f __attribute__((ext_vector_type(2)));
typedef float v8f __attribute__((ext_vector_type(8)));

// ---------------------------------------------------------------------------
// GEMM: C[M,128] = A @ W^T (+ A2 @ W2^T) + bias
//   A,A2 : [M,128] row-major (M multiple of 16)
//   W,W2 : rows are output features; row stride ldw; use cols [woff,woff+128)
// One block = 16 rows of C; 8 waves each own a 16x16 column tile.
// f32 WMMA 16x16x4, K looped 128/4 = 32 steps.
// ---------------------------------------------------------------------------
__global__ __launch_bounds__(256) void gemm_wmma_f32(
    const float* __restrict__ A, const float* __restrict__ W, int ldw, int woff,
    const float* __restrict__ A2, const float* __restrict__ W2, int ldw2, int woff2,
    const float* __restrict__ bias, float* __restrict__ C)
{
  __shared__ float lA [16 * 132];   // padded rows: stride 132 avoids bank conflicts
  __shared__ float lA2[16 * 132];

  const int tid  = threadIdx.x;
  const int wave = tid >> 5;
  const int lane = tid & 31;
  const int r0   = blockIdx.x * 16;
  const int c0   = wave * 16;

  // Stage A tile: rows r0..r0+15 of a [*,128] matrix are one contiguous 2048-float span.
  const float* Ab = A + (size_t)r0 * DIM;
  for (int i = tid; i < 16 * DIM; i += 256)
    lA[(i >> 7) * 132 + (i & 127)] = Ab[i];
  if (A2) {
    const float* A2b = A2 + (size_t)r0 * DIM;
    for (int i = tid; i < 16 * DIM; i += 256)
      lA2[(i >> 7) * 132 + (i & 127)] = A2b[i];
  }
  __syncthreads();

  // f32 WMMA fragment mapping (ISA 7.12.2):
  //  A 16x4:  lanes 0-15 hold M=lane, K=k+{0,1}; lanes 16-31 hold M=lane-16, K=k+{2,3}
  //  B 4x16:  lanes 0-15 hold N=lane, K=k+{0,1}; lanes 16-31 hold N=lane-16, K=k+{2,3}
  const int mrow = lane & 15;
  const int koff = (lane >> 4) << 1;            // 0 or 2
  const int bcol = c0 + mrow;                   // output feature (row of W)

  v8f acc = {};
  const float* Wb = W + (size_t)bcol * ldw + woff + koff;
#pragma unroll
  for (int k = 0; k < DIM; k += 4) {
    v2f a = *(const v2f*)&lA[mrow * 132 + k + koff];
    v2f b = *(const v2f*)&Wb[k];
    acc = __builtin_amdgcn_wmma_f32_16x16x4_f32(false, a, false, b,
                                                (short)0, acc, false, false);
  }
  if (A2) {
    const float* W2b = W2 + (size_t)bcol * ldw2 + woff2 + koff;
#pragma unroll
    for (int k = 0; k < DIM; k += 4) {
      v2f a = *(const v2f*)&lA2[mrow * 132 + k + koff];
      v2f b = *(const v2f*)&W2b[k];
      acc = __builtin_amdgcn_wmma_f32_16x16x4_f32(false, a, false, b,
                                                  (short)0, acc, false, false);
    }
  }

  // C/D layout: VGPR v -> M = v (lanes 0-15) / v+8 (lanes 16-31), N = lane&15
  const float bj    = bias ? bias[c0 + mrow] : 0.0f;
  const int   rbase = r0 + ((lane >> 4) << 3);
#pragma unroll
  for (int v = 0; v < 8; ++v)
    C[(size_t)(rbase + v) * DIM + c0 + mrow] = acc[v] + bj;
}

// ---------------------------------------------------------------------------
// BatchNorm (training-mode batch stats) helpers
// ---------------------------------------------------------------------------
__global__ void bn_partial(const float* __restrict__ x, double* __restrict__ part)
{
  const int j     = threadIdx.x;                      // column, blockDim.x == 128
  const int chunk = (N_NODES + BN_PB - 1) / BN_PB;    // 200
  const int r0    = blockIdx.x * chunk;
  const int r1    = (r0 + chunk < N_NODES) ? (r0 + chunk) : N_NODES;
  double s = 0.0, sq = 0.0;
  for (int r = r0; r < r1; ++r) {
    float v = x[(size_t)r * DIM + j];
    s  += v;
    sq += (double)v * (double)v;
  }
  part[(size_t)blockIdx.x * 256 + j]       = s;
  part[(size_t)blockIdx.x * 256 + 128 + j] = sq;
}

__global__ void bn_finalize(const double* __restrict__ part,
                            const float* __restrict__ g, const float* __restrict__ be,
                            float* __restrict__ ac)
{
  const int j = threadIdx.x;                          // 128 threads
  double s = 0.0, sq = 0.0;
  for (int b = 0; b < BN_PB; ++b) {
    s  += part[(size_t)b * 256 + j];
    sq += part[(size_t)b * 256 + 128 + j];
  }
  double m   = s / (double)N_NODES;
  double var = sq / (double)N_NODES - m * m;          // biased variance
  float inv  = rsqrtf((float)var + 1e-5f);
  float a    = g[j] * inv;
  ac[j]       = a;
  ac[DIM + j] = be[j] - (float)m * a;
}

__global__ void bn_apply_relu(const float* __restrict__ x, const float* __restrict__ ac,
                              float* __restrict__ y)
{
  int i = blockIdx.x * blockDim.x + threadIdx.x;
  if (i >= N_NODES * DIM) return;
  int j = i & 127;
  float v = fmaf(x[i], ac[j], ac[DIM + j]);
  y[i] = v > 0.0f ? v : 0.0f;
}

// ---------------------------------------------------------------------------
// Edge kernels
// ---------------------------------------------------------------------------
__global__ void zero_f32(float* __restrict__ p, int n)
{
  int i = blockIdx.x * blockDim.x + threadIdx.x;
  if (i < n) p[i] = 0.0f;
}

__global__ void deg_count(const int* __restrict__ dst, float* __restrict__ deg)
{
  int e = blockIdx.x * blockDim.x + threadIdx.x;
  if (e < N_EDGES) unsafeAtomicAdd(&deg[dst[e]], 1.0f);   // integer counts: exact in f32
}

// aggr = deg * Ai + skip + bconv   (segment_sum(Ai[dst]) == deg * Ai)
__global__ void aggr_init(const float* __restrict__ Ai, const float* __restrict__ skip,
                          const float* __restrict__ deg, const float* __restrict__ bconv,
                          float* __restrict__ aggr)
{
  int i = blockIdx.x * blockDim.x + threadIdx.x;
  if (i >= N_NODES * DIM) return;
  int n = i >> 7, j = i & 127;
  aggr[i] = fmaf(deg[n], Ai[i], skip[i] + bconv[j]);
}

// aggr[dst] += Bj[src]   (coalesced: consecutive lanes = consecutive d of one edge)
__global__ void scatter_msg(const float* __restrict__ Bj, const int* __restrict__ src,
                            const int* __restrict__ dst, float* __restrict__ aggr)
{
  const long long total = (long long)N_EDGES * DIM;
  for (long long i = (long long)blockIdx.x * blockDim.x + threadIdx.x; i < total;
       i += (long long)gridDim.x * blockDim.x) {
    int e = (int)(i >> 7), j = (int)(i & 127);
    unsafeAtomicAdd(&aggr[(size_t)dst[e] * DIM + j], Bj[(size_t)src[e] * DIM + j]);
  }
}

// ---------------------------------------------------------------------------
// GRU pointwise
// ---------------------------------------------------------------------------
__global__ void gru_rh(float* __restrict__ rlin, const float* __restrict__ Hi)
{
  int i = blockIdx.x * blockDim.x + threadIdx.x;
  if (i >= N_NODES * DIM) return;
  float r = 1.0f / (1.0f + expf(-rlin[i]));
  rlin[i] = r * Hi[i];
}

__global__ void gru_final(const float* __restrict__ zlin, const float* __restrict__ hlin,
                          const float* __restrict__ Hi, float* __restrict__ h)
{
  int i = blockIdx.x * blockDim.x + threadIdx.x;
  if (i >= N_NODES * DIM) return;
  float z  = 1.0f / (1.0f + expf(-zlin[i]));
  float ht = tanhf(hlin[i]);
  h[i] = z * Hi[i] + (1.0f - z) * ht;
}

// ---------------------------------------------------------------------------
// LinkDecoder: out[el] = sum_d h[s,d]*h[t,d]*Wdec[d] + bdec   (one wave / pair)
// ---------------------------------------------------------------------------
__global__ void decoder(const float* __restrict__ h, const int* __restrict__ eli,
                        const float* __restrict__ Wdec, const float* __restrict__ bdec,
                        float* __restrict__ out)
{
  int gt   = blockIdx.x * blockDim.x + threadIdx.x;
  int el   = gt >> 5;
  int lane = gt & 31;
  if (el >= N_EL) return;
  size_t s = (size_t)eli[el] * DIM;
  size_t t = (size_t)eli[N_EL + el] * DIM;
  float acc = 0.0f;
#pragma unroll
  for (int d0 = 0; d0 < DIM; d0 += 32) {
    int d = d0 + lane;
    acc += h[s + d] * h[t + d] * Wdec[d];
  }
  for (int off = 16; off; off >>= 1) acc += __shfl_xor(acc, off, 32);
  if (lane == 0) out[el] = acc + bdec[0];
}

// ---------------------------------------------------------------------------
extern "C" void kernel_launch(void* const* d_in, const int* in_sizes, int n_in,
                              void* d_out, int out_size, void* d_ws, size_t ws_size,
                              hipStream_t stream)
{
  const float* x      = (const float*)d_in[0];
  const int*   ei     = (const int*)  d_in[1];
  const int*   eli    = (const int*)  d_in[2];
  const float* H      = (const float*)d_in[3];
  const float* W_mlp  = (const float*)d_in[4];
  const float* b_mlp  = (const float*)d_in[5];
  const float* g_mlp  = (const float*)d_in[6];
  const float* be_mlp = (const float*)d_in[7];
  const float* Wskip  = (const float*)d_in[8];
  const float* bskip  = (const float*)d_in[9];
  const float* Wmsg   = (const float*)d_in[10];
  const float* bconv  = (const float*)d_in[11];
  const float* g_bn   = (const float*)d_in[12];
  const float* be_bn  = (const float*)d_in[13];
  const float* Wz     = (const float*)d_in[14];
  const float* bz     = (const float*)d_in[15];
  const float* Wr     = (const float*)d_in[16];
  const float* br     = (const float*)d_in[17];
  const float* Wh     = (const float*)d_in[18];
  const float* bh     = (const float*)d_in[19];
  const float* Wdec   = (const float*)d_in[20];
  const float* bdec   = (const float*)d_in[21];
  float* out = (float*)d_out;

  const size_t ND = (size_t)N_NODES * DIM;          // 6.4M floats
  char* ws = (char*)d_ws;
  float*  h    = (float*)(ws);
  float*  t0   = (float*)(ws +  4 * ND);
  float*  t1   = (float*)(ws +  8 * ND);
  float*  t2   = (float*)(ws + 12 * ND);
  float*  t3   = (float*)(ws + 16 * ND);
  float*  deg  = (float*)(ws + 20 * ND);
  double* part = (double*)(ws + 20 * ND + (((size_t)N_NODES * 4 + 7) & ~(size_t)7));
  float*  ac   = (float*)((char*)part + (size_t)BN_PB * 256 * 8);

  const int* esrc = ei;
  const int* edst = ei + N_EDGES;

  const int gemm_grid = N_NODES / 16;               // 3125 (exact)
  const int ew_grid   = (int)((ND + 255) / 256);    // 25000

  // degree (reused by both layers)
  zero_f32 <<<(N_NODES + 255) / 256, 256, 0, stream>>>(deg, N_NODES);
  deg_count<<<(N_EDGES + 255) / 256, 256, 0, stream>>>(edst, deg);

  // mlp_transform: Linear + BN + ReLU -> h
  gemm_wmma_f32<<<gemm_grid, 256, 0, stream>>>(x, W_mlp, DIM, 0,
                                               nullptr, nullptr, 0, 0, b_mlp, t0);
  bn_partial   <<<BN_PB, DIM, 0, stream>>>(t0, part);
  bn_finalize  <<<1, DIM, 0, stream>>>(part, g_mlp, be_mlp, ac);
  bn_apply_relu<<<ew_grid, 256, 0, stream>>>(t0, ac, h);

  for (int i = 0; i < N_LAYERS; ++i) {
    const float* Hi     = H    + (size_t)i * ND;
    const float* Wmsg_i = Wmsg + (size_t)i * DIM * 2 * DIM;
    const float* Wz_i   = Wz   + (size_t)i * DIM * 2 * DIM;
    const float* Wr_i   = Wr   + (size_t)i * DIM * 2 * DIM;
    const float* Wh_i   = Wh   + (size_t)i * DIM * 2 * DIM;

    // skip -> t0 ; Ai -> t1 ; Bj -> t2
    gemm_wmma_f32<<<gemm_grid, 256, 0, stream>>>(h, Wskip + (size_t)i * DIM * DIM, DIM, 0,
                                                 nullptr, nullptr, 0, 0, bskip + i * DIM, t0);
    gemm_wmma_f32<<<gemm_grid, 256, 0, stream>>>(h, Wmsg_i, 2 * DIM, 0,
                                                 nullptr, nullptr, 0, 0, nullptr, t1);
    gemm_wmma_f32<<<gemm_grid, 256, 0, stream>>>(h, Wmsg_i, 2 * DIM, DIM,
                                                 nullptr, nullptr, 0, 0, nullptr, t2);

    // aggr = deg*Ai + skip + bconv, then += scatter of Bj[src] into dst
    aggr_init  <<<ew_grid, 256, 0, stream>>>(t1, t0, deg, bconv + i * DIM, t3);
    scatter_msg<<<8192, 256, 0, stream>>>(t2, esrc, edst, t3);

    // BN + ReLU -> out (t0)
    bn_partial   <<<BN_PB, DIM, 0, stream>>>(t3, part);
    bn_finalize  <<<1, DIM, 0, stream>>>(part, g_bn + i * DIM, be_bn + i * DIM, ac);
    bn_apply_relu<<<ew_grid, 256, 0, stream>>>(t3, ac, t0);

    // GRU: Zlin -> t1, Rlin -> t2 (concat GEMMs split over woff)
    gemm_wmma_f32<<<gemm_grid, 256, 0, stream>>>(t0, Wz_i, 2 * DIM, 0,
                                                 Hi, Wz_i, 2 * DIM, DIM, bz + i * DIM, t1);
    gemm_wmma_f32<<<gemm_grid, 256, 0, stream>>>(t0, Wr_i, 2 * DIM, 0,
                                                 Hi, Wr_i, 2 * DIM, DIM, br + i * DIM, t2);
    gru_rh<<<ew_grid, 256, 0, stream>>>(t2, Hi);   // t2 = sigmoid(Rlin) * Hi
    gemm_wmma_f32<<<gemm_grid, 256, 0, stream>>>(t0, Wh_i, 2 * DIM, 0,
                                                 t2, Wh_i, 2 * DIM, DIM, bh + i * DIM, t3);
    gru_final<<<ew_grid, 256, 0, stream>>>(t1, t3, Hi, h);  // h = z*Hi + (1-z)*tanh
  }

  decoder<<<(N_EL * 32 + 255) / 256, 256, 0, stream>>>(h, eli, Wdec, bdec, out);
}